// SparseTopKMoE_13159779795307
// MI455X (gfx1250) — compile-verified
//
#include <hip/hip_runtime.h>
#include <math.h>

// ---------------------------------------------------------------------------
// Sparse top-2 MoE, fp32, WMMA f32_16x16x4 path for gfx1250 (wave32).
// N=8192 tokens, C=H=O=1024, E=16 experts, top-k=2.
// M=32 token tiles: each weight (B) fragment feeds 2 WMMAs.
// x-row gather uses GLOBAL_LOAD_ASYNC_TO_LDS_B128 (ASYNCcnt-tracked DMA into
// LDS, no VGPR staging).
// ---------------------------------------------------------------------------

typedef __attribute__((ext_vector_type(2))) float v2f;
typedef __attribute__((ext_vector_type(8))) float v8f;

constexpr int N_TOK = 8192;
constexpr int CDIM  = 1024;
constexpr int HDIM  = 1024;
constexpr int ODIM  = 1024;
constexpr int NEXP  = 16;
constexpr int MTILE = 32;
constexpr int LDS_STRIDE = 1032;   // 1024 + 8: bank spread, keeps 16B alignment

// ---------------------------------------------------------------------------
// Kernel 1: zero output + router counters
// ---------------------------------------------------------------------------
__global__ __launch_bounds__(256) void moe_zero_kernel(float4* out4, int n4, int* counts)
{
    int i = blockIdx.x * 256 + threadIdx.x;
    if (i < n4) {
        float4 z; z.x = 0.f; z.y = 0.f; z.z = 0.f; z.w = 0.f;
        out4[i] = z;
    }
    if (i < NEXP) counts[i] = 0;
}

// ---------------------------------------------------------------------------
// Kernel 2: router. One wave (32 lanes) per token.
//   logits[e] = x[n,:] . Wr[:,e] + br[e]
//   top-2 (first-index wins ties, matching jax.lax.top_k), pairwise softmax.
// ---------------------------------------------------------------------------
__global__ __launch_bounds__(256) void moe_router_kernel(
    const float* __restrict__ x, const float* __restrict__ Wr,
    const float* __restrict__ br,
    int* __restrict__ counts, int* __restrict__ tok_e, float* __restrict__ tok_w)
{
    const int lane = threadIdx.x & 31;
    const int wave = threadIdx.x >> 5;
    const int n = blockIdx.x * 8 + wave;
    if (n >= N_TOK) return;

    float acc[NEXP];
#pragma unroll
    for (int e = 0; e < NEXP; ++e) acc[e] = 0.f;

    const float* xr = x + (size_t)n * CDIM;
    for (int c = lane; c < CDIM; c += 32) {
        float xv = xr[c];
        const float* wr = Wr + (size_t)c * NEXP;
#pragma unroll
        for (int e = 0; e < NEXP; ++e) acc[e] += xv * wr[e];
    }
    // full-wave butterfly reduce: every lane ends with the full sums
#pragma unroll
    for (int e = 0; e < NEXP; ++e) {
        float v = acc[e];
        for (int off = 16; off; off >>= 1) v += __shfl_xor(v, off, 32);
        acc[e] = v;
    }
    if (lane == 0) {
        float l0 = -3.4e38f, l1 = -3.4e38f;
        int i0 = 0, i1 = 0;
#pragma unroll
        for (int e = 0; e < NEXP; ++e) {
            float le = acc[e] + br[e];
            if (le > l0) { l0 = le; i0 = e; }   // strict '>' => first index wins
        }
#pragma unroll
        for (int e = 0; e < NEXP; ++e) {
            if (e == i0) continue;
            float le = acc[e] + br[e];
            if (le > l1) { l1 = le; i1 = e; }
        }
        // renormalized top-2 of softmax == softmax over the two logits
        float ed = expf(l1 - l0);            // l1 <= l0 -> stable
        float w0 = 1.0f / (1.0f + ed);
        float w1 = ed * w0;
        tok_e[2 * n]     = i0;
        tok_e[2 * n + 1] = i1;
        tok_w[2 * n]     = w0;
        tok_w[2 * n + 1] = w1;
        atomicAdd(&counts[i0], 1);
        atomicAdd(&counts[i1], 1);
    }
}

// ---------------------------------------------------------------------------
// Kernel 3: exclusive scan over 16 counts (single thread; trivial)
// ---------------------------------------------------------------------------
__global__ void moe_scan_kernel(const int* __restrict__ counts,
                                int* __restrict__ offs, int* __restrict__ fill)
{
    if (threadIdx.x == 0 && blockIdx.x == 0) {
        int s = 0;
        for (int e = 0; e < NEXP; ++e) { offs[e] = s; fill[e] = s; s += counts[e]; }
    }
}

// ---------------------------------------------------------------------------
// Kernel 4: scatter tokens into expert-grouped entry lists
// ---------------------------------------------------------------------------
__global__ __launch_bounds__(256) void moe_scatter_kernel(
    const int* __restrict__ tok_e, const float* __restrict__ tok_w,
    int* __restrict__ fill, int* __restrict__ perm, float* __restrict__ gatew)
{
    int n = blockIdx.x * 256 + threadIdx.x;
    if (n >= N_TOK) return;
#pragma unroll
    for (int k = 0; k < 2; ++k) {
        int e = tok_e[2 * n + k];
        int pos = atomicAdd(&fill[e], 1);
        perm[pos]  = n;
        gatew[pos] = tok_w[2 * n + k];
    }
}

// ---------------------------------------------------------------------------
// WMMA tile GEMM: acc[32 x 128] += LDS_A[32 x 1024] * B[1024 x 1024] slice.
// A fragment (16x4 f32): lane holds row M=lane&15; VGPRs hold K = k0+2*(lane>>4)+{0,1}.
// B fragment (4x16 f32): lane holds col N=lane&15; VGPRs hold K rows likewise.
// Two M sub-tiles (rows 0-15, 16-31) share every B fragment.
// ---------------------------------------------------------------------------
__device__ __forceinline__ v8f wmma4(v2f a, v2f b, v8f c)
{
    return __builtin_amdgcn_wmma_f32_16x16x4_f32(
        false, a, false, b, (short)0, c, false, false);
}

__device__ __forceinline__ void gemm_32x1024(
    const float* __restrict__ ldsA, const float* __restrict__ B,
    int lane, int colbase, v8f acc[2][8])
{
    const int r  = lane & 15;
    const int kk = (lane >> 4) * 2;          // 0 or 2
    for (int k0 = 0; k0 < 1024; k0 += 4) {
        const float* ar0 = ldsA + r * LDS_STRIDE + k0 + kk;
        const float* ar1 = ar0 + 16 * LDS_STRIDE;
        v2f a0; a0.x = ar0[0]; a0.y = ar0[1];
        v2f a1; a1.x = ar1[0]; a1.y = ar1[1];
        const float* Bk = B + (size_t)(k0 + kk) * 1024;
        __builtin_prefetch(Bk + 16 * 1024, 0, 1);   // stream weights ahead
#pragma unroll
        for (int t = 0; t < 8; ++t) {
            const int n = colbase + t * 16 + r;
            v2f b; b.x = Bk[n]; b.y = Bk[1024 + n];
            acc[0][t] = wmma4(a0, b, acc[0][t]);
            acc[1][t] = wmma4(a1, b, acc[1][t]);
        }
    }
}

// ---------------------------------------------------------------------------
// Kernel 5: fused per-(expert, 32-token tile) MLP:
//   h = LN(x W1 + b1; g1, be1); h = gelu(h); y += gate * (h W2 + b2)
// 256 threads = 8 waves; wave w owns hidden/out columns [128w, 128w+128).
// ---------------------------------------------------------------------------
__global__ __launch_bounds__(256) void moe_expert_kernel(
    const float* __restrict__ x,
    const float* __restrict__ W1, const float* __restrict__ b1,
    const float* __restrict__ g1, const float* __restrict__ be1,
    const float* __restrict__ W2, const float* __restrict__ b2,
    const int* __restrict__ counts, const int* __restrict__ offs,
    const int* __restrict__ perm, const float* __restrict__ gatew,
    float* __restrict__ out)
{
    extern __shared__ float smem[];               // MTILE x LDS_STRIDE floats
    const int e      = blockIdx.y;
    const int cnt    = counts[e];
    const int tile0  = blockIdx.x * MTILE;
    if (tile0 >= cnt) return;
    const int base = offs[e];

    const int tid  = threadIdx.x;
    const int lane = tid & 31;
    const int wave = tid >> 5;
    const int colbase = wave * 128;

    // ---- gather 32 token rows of x into LDS via async memory->LDS DMA ------
    // (padded rows clamp to last valid entry; 16B aligned on both sides;
    //  flat->LDS aperture truncation makes low 32 addr bits the LDS offset)
    for (int i = tid; i < MTILE * 256; i += 256) { // 256 float4 per 1024-f row
        int row = i >> 8;
        int c4  = (i & 255) << 2;
        int entry = tile0 + row;
        if (entry >= cnt) entry = cnt - 1;
        int token = perm[base + entry];
        const float* gp = x + (size_t)token * CDIM + c4;
        unsigned int       ldsoff = (unsigned int)(uintptr_t)(smem + row * LDS_STRIDE + c4);
        unsigned long long gaddr  = (unsigned long long)(uintptr_t)gp;
        asm volatile("global_load_async_to_lds_b128 %0, %1, off"
                     :: "v"(ldsoff), "v"(gaddr) : "memory");
    }
    asm volatile("s_wait_asynccnt 0" ::: "memory");   // drain this wave's DMAs
    __syncthreads();                                  // publish LDS to all waves

    // ---- GEMM1: h = x @ W1[e] ----
    v8f acc[2][8];
#pragma unroll
    for (int m = 0; m < 2; ++m)
#pragma unroll
        for (int t = 0; t < 8; ++t)
#pragma unroll
            for (int v = 0; v < 8; ++v) acc[m][t][v] = 0.f;
    gemm_32x1024(smem, W1 + (size_t)e * CDIM * HDIM, lane, colbase, acc);
    __syncthreads();                               // done reading x from LDS

    // ---- bias, write h back over LDS ----
    const int r  = lane & 15;
    const int mh = (lane >> 4) * 8;                // row base for C/D frags
#pragma unroll
    for (int t = 0; t < 8; ++t) {
        const int n  = colbase + t * 16 + r;
        const float bb = b1[e * HDIM + n];
#pragma unroll
        for (int v = 0; v < 8; ++v) {
            smem[(mh + v) * LDS_STRIDE + n]        = acc[0][t][v] + bb;
            smem[(16 + mh + v) * LDS_STRIDE + n]   = acc[1][t][v] + bb;
        }
    }
    __syncthreads();

    // ---- LayerNorm + exact GELU, in place. 8 threads per row (32 rows). ----
    {
        const int row = tid >> 3;                  // 0..31
        const int j   = tid & 7;
        float* hrow = smem + row * LDS_STRIDE;
        float s = 0.f;
        for (int c = j; c < HDIM; c += 8) s += hrow[c];
        for (int off = 4; off; off >>= 1) s += __shfl_xor(s, off, 32);
        const float mu = s * (1.0f / (float)HDIM);
        float q = 0.f;
        for (int c = j; c < HDIM; c += 8) { float d = hrow[c] - mu; q += d * d; }
        for (int off = 4; off; off >>= 1) q += __shfl_xor(q, off, 32);
        const float rs = rsqrtf(q * (1.0f / (float)HDIM) + 1e-5f);
        for (int c = j; c < HDIM; c += 8) {
            float hv = (hrow[c] - mu) * rs * g1[e * HDIM + c] + be1[e * HDIM + c];
            hrow[c] = hv * 0.5f * (1.0f + erff(hv * 0.70710678118654752f));
        }
    }
    __syncthreads();

    // ---- GEMM2: y = h @ W2[e] ----
    v8f acc2[2][8];
#pragma unroll
    for (int m = 0; m < 2; ++m)
#pragma unroll
        for (int t = 0; t < 8; ++t)
#pragma unroll
            for (int v = 0; v < 8; ++v) acc2[m][t][v] = 0.f;
    gemm_32x1024(smem, W2 + (size_t)e * HDIM * ODIM, lane, colbase, acc2);

    // ---- gated accumulate into out (exactly 2 commutative adds per element) -
#pragma unroll
    for (int t = 0; t < 8; ++t) {
        const int n  = colbase + t * 16 + r;
        const float bb = b2[e * ODIM + n];
#pragma unroll
        for (int v = 0; v < 8; ++v) {
#pragma unroll
            for (int m = 0; m < 2; ++m) {
                const int entry = tile0 + m * 16 + mh + v;
                if (entry < cnt) {
                    const int gi = base + entry;
                    const int token = perm[gi];
                    const float w = gatew[gi];
                    unsafeAtomicAdd(out + (size_t)token * ODIM + n,
                                    w * (acc2[m][t][v] + bb));
                }
            }
        }
    }
}

// ---------------------------------------------------------------------------
// Host launcher
// ---------------------------------------------------------------------------
extern "C" void kernel_launch(void* const* d_in, const int* in_sizes, int n_in,
                              void* d_out, int out_size, void* d_ws, size_t ws_size,
                              hipStream_t stream)
{
    const float* x   = (const float*)d_in[0];
    const float* Wr  = (const float*)d_in[1];
    const float* br  = (const float*)d_in[2];
    const float* W1  = (const float*)d_in[3];
    const float* b1  = (const float*)d_in[4];
    const float* g1  = (const float*)d_in[5];
    const float* be1 = (const float*)d_in[6];
    const float* W2  = (const float*)d_in[7];
    const float* b2  = (const float*)d_in[8];
    float* out = (float*)d_out;

    // workspace layout (262,400 bytes total)
    int*   ws_i   = (int*)d_ws;
    float* ws_f   = (float*)d_ws;
    int*   counts = ws_i;                     // [16]
    int*   offs   = ws_i + 16;                // [16]
    int*   fill   = ws_i + 32;                // [16]
    int*   tok_e  = ws_i + 64;                // [2N]
    int*   perm   = ws_i + 64 + 2 * N_TOK;    // [2N]
    float* tok_w  = ws_f + 64 + 4 * N_TOK;    // [2N]
    float* gatew  = ws_f + 64 + 6 * N_TOK;    // [2N]

    // 1) zero out + counters
    const int n4 = N_TOK * ODIM / 4;
    moe_zero_kernel<<<(n4 + 255) / 256, 256, 0, stream>>>((float4*)out, n4, counts);

    // 2) router (wave per token)
    moe_router_kernel<<<N_TOK / 8, 256, 0, stream>>>(x, Wr, br, counts, tok_e, tok_w);

    // 3) scan
    moe_scan_kernel<<<1, 32, 0, stream>>>(counts, offs, fill);

    // 4) scatter into expert-grouped lists
    moe_scatter_kernel<<<(N_TOK + 255) / 256, 256, 0, stream>>>(tok_e, tok_w, fill, perm, gatew);

    // 5) fused expert MLP; grid = (max tiles per expert, experts)
    dim3 grid(N_TOK / MTILE, NEXP);           // 256 x 16; idle tiles exit early
    size_t smem = (size_t)MTILE * LDS_STRIDE * sizeof(float);   // 132,096 B
    moe_expert_kernel<<<grid, 256, smem, stream>>>(
        x, W1, b1, g1, be1, W2, b2, counts, offs, perm, gatew, out);
}